// G_SAE_8907762172112
// MI455X (gfx1250) — compile-verified
//
#include <hip/hip_runtime.h>
#include <hip/hip_bf16.h>
#include <math.h>

// ---------------------------------------------------------------------------
// SAE forward on gfx1250 (MI455X):
//   latents_pre = x @ W_enc^T + b_enc     (fp32 WMMA 16x16x4, TDM-staged LDS)
//   latents     = topk_sigmoid(latents_pre, k=410)  (radix-select per row)
//   recons      = latents @ W_dec^T + b_dec (same GEMM kernel)
// ---------------------------------------------------------------------------

typedef float v2f __attribute__((ext_vector_type(2)));
typedef float v8f __attribute__((ext_vector_type(8)));
typedef unsigned int v4u __attribute__((ext_vector_type(4)));
typedef int v8i __attribute__((ext_vector_type(8)));
typedef int v4i __attribute__((ext_vector_type(4)));

#define BATCH     16384
#define NFEAT     1024
#define NLAT      4096
#define TOPK      410

#define KC        32          // K-block staged in LDS
#define LDS_PAD   4           // stride 36 dwords (TDM pad: +4 dw after every 32 dw)
#define TILE_M    64          // workgroup M tile (4 waves along M)
#define TILE_N    128         // workgroup N tile (2 waves along N, 64 each)

// ---------------------------------------------------------------------------
// Issue one 2D TDM tile load: rows x cols (fp32) from row-major global
// (row stride = strideElems) into LDS at byte offset ldsOff, padding +4
// DWORDs after every 32 DWORDs stored (-> LDS row stride 36 dwords).
// ---------------------------------------------------------------------------
__device__ __forceinline__ void tdm_load_2d(unsigned ldsOff, const void* gptr,
                                            unsigned cols, unsigned rows,
                                            unsigned tensorD0, unsigned tensorD1,
                                            unsigned strideElems) {
    unsigned long long ga = (unsigned long long)(uintptr_t)gptr;
    v4u g0;
    g0[0] = 1u;                                            // count=1 (valid), user mode
    g0[1] = ldsOff;                                        // lds_addr (bytes)
    g0[2] = (unsigned)(ga & 0xFFFFFFFFu);                  // global_addr[31:0]
    g0[3] = ((unsigned)(ga >> 32) & 0x01FFFFFFu)           // global_addr[56:32]
            | 0x80000000u;                                 // type=2 ("image")
    v8i g1;
    g1[0] = (int)((2u << 16)                               // data_size = 4B
                | (1u << 20)                               // pad_enable
                | (4u << 22)                               // pad_interval: 32 DWORDs
                | (3u << 25));                             // pad_amount: 4 DWORDs
    g1[1] = (int)((tensorD0 & 0xFFFFu) << 16);             // tensor_dim0[15:0]
    g1[2] = (int)((tensorD0 >> 16) | ((tensorD1 & 0xFFFFu) << 16));
    g1[3] = (int)((tensorD1 >> 16) | ((cols & 0xFFFFu) << 16));  // tile_dim0
    g1[4] = (int)(rows & 0xFFFFu);                         // tile_dim1 (tile_dim2=0)
    g1[5] = (int)strideElems;                              // tensor_dim0_stride[31:0]
    g1[6] = 0;                                             // stride hi / dim1_stride lo
    g1[7] = 0;
    v4i z4 = {0, 0, 0, 0};                                 // groups 2/3: unused dims = 0
    v8i z8 = {0, 0, 0, 0, 0, 0, 0, 0};
    __builtin_amdgcn_tensor_load_to_lds(g0, g1, z4, z4, z8, 0);
}

// C[M,N] = A[M,K] * W[N,K]^T + bias[N]   (NT GEMM, both operands K-contiguous)
__global__ __launch_bounds__(256)
void sae_gemm_nt_f32(float* __restrict__ C,
                     const float* __restrict__ A,
                     const float* __restrict__ W,
                     const float* __restrict__ bias,
                     int M, int N, int K) {
    __shared__ float As[2][TILE_M][KC + LDS_PAD];
    __shared__ float Ws[2][TILE_N][KC + LDS_PAD];

    const int tid    = threadIdx.x;
    const int lane   = tid & 31;
    const int l      = lane & 15;          // row-within-frag
    const int half   = lane >> 4;          // lane half selects K / M sub-split
    const int waveId = tid >> 5;           // 0..7
    const int mw     = waveId & 3;         // 4 waves along M
    const int nw     = waveId >> 2;        // 2 waves along N

    const int mBlk = blockIdx.y * TILE_M;  // global M tile base
    const int nBlk = blockIdx.x * TILE_N;  // global N tile base
    const int mLoc = mw * 16;              // wave M offset in tile
    const int nLoc = nw * 64;              // wave N offset in tile (4 x 16 subtiles)

    const unsigned ldsA = (unsigned)(uintptr_t)&As[0][0][0];
    const unsigned ldsW = (unsigned)(uintptr_t)&Ws[0][0][0];
    const unsigned bytesA = sizeof(As[0]);
    const unsigned bytesW = sizeof(Ws[0]);

    v8f acc[4] = {};

    // prologue: TDM-stage buffer 0, producer wave waits, barrier publishes
    if (waveId == 0) {
        tdm_load_2d(ldsA, &A[(size_t)mBlk * K], KC, TILE_M, (unsigned)K, (unsigned)(M - mBlk), (unsigned)K);
        tdm_load_2d(ldsW, &W[(size_t)nBlk * K], KC, TILE_N, (unsigned)K, (unsigned)(N - nBlk), (unsigned)K);
        __builtin_amdgcn_s_wait_tensorcnt(0);
    }
    __syncthreads();

    for (int k0 = 0; k0 < K; k0 += KC) {
        const int  buf     = (k0 / KC) & 1;
        const bool hasNext = (k0 + KC) < K;

        // async-stage next K block into the other buffer while we compute
        if (waveId == 0 && hasNext) {
            const int nb = buf ^ 1;
            tdm_load_2d(ldsA + nb * bytesA, &A[(size_t)mBlk * K + k0 + KC],
                        KC, TILE_M, (unsigned)K, (unsigned)(M - mBlk), (unsigned)K);
            tdm_load_2d(ldsW + nb * bytesW, &W[(size_t)nBlk * K + k0 + KC],
                        KC, TILE_N, (unsigned)K, (unsigned)(N - nBlk), (unsigned)K);
        }

        // ---- compute: 8 K-steps of 4, 4 N-subtiles each -> 32 WMMAs ----
        #pragma unroll
        for (int kk = 0; kk < KC; kk += 4) {
            // A frag 16x4 f32: v0 holds K=0|2 (lane halves), v1 holds K=1|3
            v2f a;
            a.x = As[buf][mLoc + l][kk + 2 * half + 0];
            a.y = As[buf][mLoc + l][kk + 2 * half + 1];
            #pragma unroll
            for (int t = 0; t < 4; ++t) {
                v2f b;
                b.x = Ws[buf][nLoc + t * 16 + l][kk + 2 * half + 0];
                b.y = Ws[buf][nLoc + t * 16 + l][kk + 2 * half + 1];
                acc[t] = __builtin_amdgcn_wmma_f32_16x16x4_f32(
                    /*neg_a=*/false, a, /*neg_b=*/false, b,
                    /*c_mod=*/(short)0, acc[t],
                    /*reuse_a=*/false, /*reuse_b=*/false);
            }
        }

        // producer waits for next buffer; barrier = (next buf ready) AND
        // (all waves done reading current buf before it is overwritten)
        if (waveId == 0 && hasNext) __builtin_amdgcn_s_wait_tensorcnt(0);
        __syncthreads();
    }

    // ---- store: C VGPR r maps to row M = r + 8*half; col N = subtile + l ----
    #pragma unroll
    for (int t = 0; t < 4; ++t) {
        const int n = nBlk + nLoc + t * 16 + l;
        const float bn = bias[n];
        #pragma unroll
        for (int r = 0; r < 8; ++r) {
            const int m = mBlk + mLoc + r + 8 * half;
            C[(size_t)m * N + n] = acc[t][r] + bn;
        }
    }
}

// ---------------------------------------------------------------------------
// Per-row top-k threshold via 4-pass byte radix-select on order-preserving
// float keys, then sigmoid-scatter. One 256-thread workgroup per row.
// ---------------------------------------------------------------------------
__device__ __forceinline__ unsigned f2key(float x) {
    unsigned u = __float_as_uint(x);
    return (u & 0x80000000u) ? ~u : (u | 0x80000000u);   // monotonic increasing
}

__global__ __launch_bounds__(256)
void sae_topk_sigmoid(const float* __restrict__ pre,
                      float* __restrict__ out,
                      int L, int k) {
    __shared__ float    row[NLAT];       // 16 KB
    __shared__ unsigned hist[256];
    __shared__ unsigned s_prefix;
    __shared__ unsigned s_k;

    const int tid = threadIdx.x;
    const size_t base = (size_t)blockIdx.x * L;

    for (int i = tid; i < L; i += 256) row[i] = pre[base + i];
    if (tid == 0) { s_prefix = 0u; s_k = (unsigned)k; }
    __syncthreads();

    // find the k-th largest key, byte by byte from the MSB
    for (int byte = 3; byte >= 0; --byte) {
        hist[tid] = 0u;
        __syncthreads();
        const unsigned pfx = s_prefix;
        const unsigned mhi = (byte == 3) ? 0u : (0xFFFFFFFFu << ((byte + 1) * 8));
        for (int i = tid; i < L; i += 256) {
            unsigned key = f2key(row[i]);
            if ((key & mhi) == (pfx & mhi))
                atomicAdd(&hist[(key >> (byte * 8)) & 255u], 1u);
        }
        __syncthreads();
        if (tid == 0) {
            unsigned rem = s_k;
            unsigned p = pfx;
            for (int b = 255; b >= 0; --b) {
                unsigned c = hist[b];
                if (c >= rem) {
                    s_prefix = p | ((unsigned)b << (byte * 8));
                    s_k = rem;
                    break;
                }
                rem -= c;
            }
        }
        __syncthreads();
    }

    const unsigned T = s_prefix;         // threshold key (k-th largest)
    for (int i = tid; i < L; i += 256) {
        float v = row[i];
        float r = 0.0f;
        if (f2key(v) >= T) r = 1.0f / (1.0f + expf(-v));
        out[base + i] = r;
    }
}

// ---------------------------------------------------------------------------
extern "C" void kernel_launch(void* const* d_in, const int* in_sizes, int n_in,
                              void* d_out, int out_size, void* d_ws, size_t ws_size,
                              hipStream_t stream) {
    const float* x     = (const float*)d_in[0];   // [B, F]
    const float* W_enc = (const float*)d_in[1];   // [L, F]
    const float* b_enc = (const float*)d_in[2];   // [L]
    const float* W_dec = (const float*)d_in[3];   // [F, L]
    const float* b_dec = (const float*)d_in[4];   // [F]

    float* latents_pre = (float*)d_out;                       // [B, L]
    float* latents     = latents_pre + (size_t)BATCH * NLAT;  // [B, L]
    float* recons      = latents     + (size_t)BATCH * NLAT;  // [B, F]

    // GEMM1: latents_pre = x @ W_enc^T + b_enc   (M=B, N=L, K=F)
    sae_gemm_nt_f32<<<dim3(NLAT / TILE_N, BATCH / TILE_M), 256, 0, stream>>>(
        latents_pre, x, W_enc, b_enc, BATCH, NLAT, NFEAT);

    // top-k(410) + sigmoid scatter, one block per row
    sae_topk_sigmoid<<<BATCH, 256, 0, stream>>>(latents_pre, latents, NLAT, TOPK);

    // GEMM2: recons = latents @ W_dec^T + b_dec  (M=B, N=F, K=L)
    sae_gemm_nt_f32<<<dim3(NFEAT / TILE_N, BATCH / TILE_M), 256, 0, stream>>>(
        recons, latents, W_dec, b_dec, BATCH, NFEAT, NLAT);
}